// CapsuleLayer_26216480375245
// MI455X (gfx1250) — compile-verified
//
#include <hip/hip_runtime.h>

// Problem constants (from reference setup_inputs)
#define BB 128           // batch
#define CT 2048          // input capsules
#define JJ 32            // output capsules (num_units)
#define UU 16            // unit dim
#define II 16            // input dim
#define NCHUNK 32        // c-chunks
#define CC (CT / NCHUNK) // 64 c's per chunk
#define CSTEP 8          // c's staged per barrier round
#define LROW 33          // padded LDS row stride (bank-conflict free)

typedef float v2f __attribute__((ext_vector_type(2)));
typedef float v4f __attribute__((ext_vector_type(4)));
typedef float v8f __attribute__((ext_vector_type(8)));

// ds_swizzle with immediate xor pattern: offset[14:10]=xor_mask, [4:0]=and_mask
template <int IMM>
__device__ __forceinline__ float swz(float p) {
    return __int_as_float(__builtin_amdgcn_ds_swizzle(__float_as_int(p), IMM));
}
#define SWZ_X1  0x041f
#define SWZ_X2  0x081f
#define SWZ_X4  0x101f
#define SWZ_X8  0x201f
#define SWZ_X16 0x401f

// Fused routing pass. Computes TRANSPOSED u_hat tiles: since A- and B-fragment
// stripings are identical, wmma(A=Wfrag, B=xfrag) yields D[u, b] with
// lane ln <-> b = b0+ln and u = v + 8*half down the 8 accumulator VGPRs.
// Routing logits b_ij = dot_u(u_hat, vsum) are then in-lane dots (vsum
// fragments hoisted out of the c-loop), softmax over j via ds_swizzle,
// weighted accumulation uses one LDS broadcast per tile.
// b_ij starts at zero and updates additively with fixed u_hat, so logits are
// linear in the accumulated v (vsum = v0, then v0+v1). use_v==0 => first
// iteration: uniform 1/32 coeff applied at flush; WMMA C chains across c.
__global__ __launch_bounds__(256)
void caps_pass(const float* __restrict__ x, const float* __restrict__ W,
               const float* __restrict__ vsum, float* __restrict__ partial,
               int use_v)
{
    const int mt    = blockIdx.x;      // 0..7   (batch tile of 16)
    const int chunk = blockIdx.y;      // 0..NCHUNK-1
    const int b0    = mt * 16;
    const int c0    = chunk * CC;

    const int tid  = threadIdx.x;      // 0..255
    const int wave = tid >> 5;         // 0..7  -> owns j tiles 4w..4w+3
    const int lane = tid & 31;
    const int half = lane >> 4;        // 0/1
    const int ln   = lane & 15;

    __shared__ float lds_a[CSTEP * 256];   // staged x tiles (8 KB)
    __shared__ float lds_b[16 * LROW];     // routing logits [b-row][j]
    __shared__ float lds_c[16 * LROW];     // softmax coeffs [b-row][j]

    // Hoisted vsum fragments: vreg[q][v] = vsum[b0+ln, 4*wave+q, v + 8*half]
    float vreg[4][8];
    if (use_v) {
        const float* vb = vsum + (size_t)(b0 + ln) * 512 + half * 8;
        #pragma unroll
        for (int q = 0; q < 4; ++q) {
            const v4f lo = *(const v4f*)(vb + (wave * 4 + q) * 16);
            const v4f hi = *(const v4f*)(vb + (wave * 4 + q) * 16 + 4);
            vreg[q][0] = lo.x; vreg[q][1] = lo.y; vreg[q][2] = lo.z; vreg[q][3] = lo.w;
            vreg[q][4] = hi.x; vreg[q][5] = hi.y; vreg[q][6] = hi.z; vreg[q][7] = hi.w;
        }
    }

    // transposed s accumulators: 4 j-tiles per wave, 8 VGPRs each
    v8f sacc[4];
    #pragma unroll
    for (int q = 0; q < 4; ++q) sacc[q] = {};

    for (int cb = 0; cb < CC; cb += CSTEP) {
        __syncthreads();
        // stage x tiles for CSTEP c's: lds_a[cl][r][i] = x[b0+r, c0+cb+cl, i]
        #pragma unroll
        for (int cl = 0; cl < CSTEP; ++cl) {
            int r = tid >> 4, i = tid & 15;
            lds_a[cl * 256 + tid] =
                x[((size_t)(b0 + r) * CT + (c0 + cb + cl)) * II + i];
        }
        __syncthreads();

        for (int cl = 0; cl < CSTEP; ++cl) {
            const int c = c0 + cb + cl;

            // x fragments (B operand): lane<16: N=b-col=ln, K={4g,4g+1};
            // lane>=16: K={4g+2,4g+3}
            v2f xfrag[4];
            #pragma unroll
            for (int g = 0; g < 4; ++g) {
                int i0 = 4 * g + 2 * half;
                xfrag[g].x = lds_a[cl * 256 + ln * II + i0];
                xfrag[g].y = lds_a[cl * 256 + ln * II + i0 + 1];
            }

            const float* Wc = W + (size_t)c * (JJ * II * UU);

            if (!use_v) {
                // iteration 0: chain accumulator across c (sT = sum_c u_hatT)
                #pragma unroll
                for (int q = 0; q < 4; ++q) {
                    const float* Wcj = Wc + (wave * 4 + q) * (II * UU);
                    #pragma unroll
                    for (int g = 0; g < 4; ++g) {
                        int i0 = 4 * g + 2 * half;
                        v2f wfrag;             // A operand: W^T, M=u=ln
                        wfrag.x = Wcj[(i0)     * UU + ln];
                        wfrag.y = Wcj[(i0 + 1) * UU + ln];
                        sacc[q] = __builtin_amdgcn_wmma_f32_16x16x4_f32(
                                    false, wfrag, false, xfrag[g],
                                    (short)0, sacc[q], false, false);
                    }
                }
            } else {
                // transposed u_hat tiles for this c
                v8f uh[4];
                #pragma unroll
                for (int q = 0; q < 4; ++q) {
                    const float* Wcj = Wc + (wave * 4 + q) * (II * UU);
                    v8f acc = {};
                    #pragma unroll
                    for (int g = 0; g < 4; ++g) {
                        int i0 = 4 * g + 2 * half;
                        v2f wfrag;
                        wfrag.x = Wcj[(i0)     * UU + ln];
                        wfrag.y = Wcj[(i0 + 1) * UU + ln];
                        acc = __builtin_amdgcn_wmma_f32_16x16x4_f32(
                                false, wfrag, false, xfrag[g],
                                (short)0, acc, false, false);
                    }
                    uh[q] = acc;
                }

                // logits: in-lane dot over u, then one xor-16 swap adds the
                // other u-half (lanes l and l^16 share the same b row)
                #pragma unroll
                for (int q = 0; q < 4; ++q) {
                    float p = 0.f;
                    #pragma unroll
                    for (int v = 0; v < 8; ++v)
                        p = fmaf(uh[q][v], vreg[q][v], p);
                    p += swz<SWZ_X16>(p);
                    if (half == 0) lds_b[ln * LROW + (wave * 4 + q)] = p;
                }
                __syncthreads();

                // parallel softmax: wave-half owns b-row m = 2*wave + half;
                // lane ln covers j = ln and j = ln + 16
                {
                    const int m = 2 * wave + half;
                    float bv0 = lds_b[m * LROW + ln];
                    float bv1 = lds_b[m * LROW + 16 + ln];
                    float mx = fmaxf(bv0, bv1);
                    mx = fmaxf(mx, swz<SWZ_X1>(mx));
                    mx = fmaxf(mx, swz<SWZ_X2>(mx));
                    mx = fmaxf(mx, swz<SWZ_X4>(mx));
                    mx = fmaxf(mx, swz<SWZ_X8>(mx));
                    float e0 = __expf(bv0 - mx);
                    float e1 = __expf(bv1 - mx);
                    float sm = e0 + e1;
                    sm += swz<SWZ_X1>(sm);
                    sm += swz<SWZ_X2>(sm);
                    sm += swz<SWZ_X4>(sm);
                    sm += swz<SWZ_X8>(sm);
                    float inv = 1.0f / sm;
                    lds_c[m * LROW + ln]      = e0 * inv;
                    lds_c[m * LROW + 16 + ln] = e1 * inv;
                }
                __syncthreads();

                // weighted accumulation: coeff uniform across a lane's VGPRs
                #pragma unroll
                for (int q = 0; q < 4; ++q) {
                    float cf = lds_c[ln * LROW + (wave * 4 + q)];
                    #pragma unroll
                    for (int v = 0; v < 8; ++v)
                        sacc[q][v] = fmaf(cf, uh[q][v], sacc[q][v]);
                }
            }
        }
    }

    // flush deterministic partials: partial[chunk][b][j][u];
    // lane ln owns row b = b0+ln, u-half = 8*half -> contiguous b128 stores
    const float scale = use_v ? 1.0f : (1.0f / 32.0f);
    float* pb = partial + ((size_t)chunk * BB + b0 + ln) * (JJ * UU) + half * 8;
    #pragma unroll
    for (int q = 0; q < 4; ++q) {
        const int j = wave * 4 + q;
        v4f s0 = { scale * sacc[q][0], scale * sacc[q][1],
                   scale * sacc[q][2], scale * sacc[q][3] };
        v4f s1 = { scale * sacc[q][4], scale * sacc[q][5],
                   scale * sacc[q][6], scale * sacc[q][7] };
        *(v4f*)(pb + j * UU)     = s0;
        *(v4f*)(pb + j * UU + 4) = s1;
    }
}

// Reduce partials over chunks (fixed order => deterministic), then squash
// over the J axis per (b,u) (reference squashes axis=1 of [B,J,U]).
// mode 0: vsum = v ; mode 1: vsum += v ; mode 2: out = v
__global__ __launch_bounds__(512)
void caps_reduce_squash(const float* __restrict__ partial,
                        float* __restrict__ vsum, float* __restrict__ out,
                        int mode)
{
    const int b = blockIdx.x;          // 0..127
    const int t = threadIdx.x;         // 0..511 : j = t>>4, u = t&15
    __shared__ float s_s[512];
    __shared__ float msq[16];

    float acc = 0.f;
    for (int ch = 0; ch < NCHUNK; ++ch)
        acc += partial[((size_t)ch * BB + b) * 512 + t];
    s_s[t] = acc;
    __syncthreads();

    if (t < 16) {                       // mag_sq over j for each u
        float m = 0.f;
        for (int j = 0; j < 32; ++j) { float z = s_s[j * 16 + t]; m += z * z; }
        msq[t] = m;
    }
    __syncthreads();

    const float m = msq[t & 15];
    const float v = (m / (1.f + m)) * (acc / (sqrtf(m + 1e-8f) + 1e-8f));

    if (mode == 0)      vsum[(size_t)b * 512 + t]  = v;
    else if (mode == 1) vsum[(size_t)b * 512 + t] += v;
    else                out[(size_t)b * 512 + t]   = v;
}

extern "C" void kernel_launch(void* const* d_in, const int* in_sizes, int n_in,
                              void* d_out, int out_size, void* d_ws, size_t ws_size,
                              hipStream_t stream) {
    (void)in_sizes; (void)n_in; (void)out_size; (void)ws_size;
    const float* x = (const float*)d_in[0];   // [128, 2048, 16]
    const float* W = (const float*)d_in[1];   // [1, 2048, 32, 16, 16]
    float* out = (float*)d_out;               // [128, 32, 16]

    float* partial = (float*)d_ws;                          // 32*128*512 f32 = 8 MB
    float* vsum    = partial + (size_t)NCHUNK * BB * 512;   // 128*512 f32 = 256 KB

    dim3 pg(8, NCHUNK);
    // it0: uniform coeffs, s0 -> v0
    caps_pass<<<pg, 256, 0, stream>>>(x, W, vsum, partial, 0);
    caps_reduce_squash<<<BB, 512, 0, stream>>>(partial, vsum, out, 0);
    // it1: logits = dot(u_hat, v0)
    caps_pass<<<pg, 256, 0, stream>>>(x, W, vsum, partial, 1);
    caps_reduce_squash<<<BB, 512, 0, stream>>>(partial, vsum, out, 1);
    // it2: logits = dot(u_hat, v0+v1)
    caps_pass<<<pg, 256, 0, stream>>>(x, W, vsum, partial, 1);
    caps_reduce_squash<<<BB, 512, 0, stream>>>(partial, vsum, out, 2);
}